// PMEElectrostatics_56805237457292
// MI455X (gfx1250) — compile-verified
//
#include <hip/hip_runtime.h>
#include <math.h>

#define ALPHA   0.3f
#define CUTOFF  10.0f
#define KMAX    8
#define KDIM    (2*KMAX+1)                 // 17
#define NKTOT   (KDIM*KDIM*KDIM)           // 4913
#define KZERO   (KMAX*KDIM*KDIM + KMAX*KDIM + KMAX)  // index of (0,0,0) = 2456
#define NKH     KZERO                      // half-space k count (Hermitian symmetry)
#define TWO_PI  6.28318530717958647692f
#define INV_SQRT_PI 0.564189583547756287f

#define ATOM_CHUNK 256                     // atoms per recip block (y-dim)

// ws layout (floats): [0]=real+self, [1]=recip energy,
//                     [2 .. 2+NKH) = s_re, [2+NKH .. 2+2*NKH) = s_im
#define WS_SRE 2
#define WS_SIM (2 + NKH)
#define WS_TOTAL (2 + 2*NKH)

typedef float v2f __attribute__((ext_vector_type(2)));
typedef float v8f __attribute__((ext_vector_type(8)));

__device__ __forceinline__ void invert3x3(const float* __restrict__ c,
                                          float inv[3][3], float* detOut) {
  float a00=c[0],a01=c[1],a02=c[2];
  float a10=c[3],a11=c[4],a12=c[5];
  float a20=c[6],a21=c[7],a22=c[8];
  float det = a00*(a11*a22-a12*a21) - a01*(a10*a22-a12*a20) + a02*(a10*a21-a11*a20);
  float id = 1.0f/det;
  inv[0][0] =  (a11*a22-a12*a21)*id;
  inv[0][1] = -(a01*a22-a02*a21)*id;
  inv[0][2] =  (a01*a12-a02*a11)*id;
  inv[1][0] = -(a10*a22-a12*a20)*id;
  inv[1][1] =  (a00*a22-a02*a20)*id;
  inv[1][2] = -(a00*a12-a02*a10)*id;
  inv[2][0] =  (a10*a21-a11*a20)*id;
  inv[2][1] = -(a00*a21-a01*a20)*id;
  inv[2][2] =  (a00*a11-a01*a10)*id;
  *detOut = det;
}

// ---------------------------------------------------------------------------
// Real-space sum + self energy.  2D grid: (i-tile, j-chunk), 256x256 pairs
// per block, j-atoms staged in LDS.  Uniform-branch fast path for diagonal
// cells (the common case); general triclinic min-image otherwise.
// ---------------------------------------------------------------------------
__global__ void ewald_real(const float* __restrict__ pos,
                           const float* __restrict__ q,
                           const float* __restrict__ cell,
                           float* __restrict__ partial, int n) {
  __shared__ float4 tile[256];
  __shared__ float  red[256];
  const int tid = threadIdx.x;
  const int i = blockIdx.x * 256 + tid;
  const int jbase = blockIdx.y * 256;

  const bool diag = (cell[1]==0.f && cell[2]==0.f && cell[3]==0.f &&
                     cell[5]==0.f && cell[6]==0.f && cell[7]==0.f);

  float xi=0.f, yi=0.f, zi=0.f, qi=0.f;
  if (i < n) { xi=pos[3*i]; yi=pos[3*i+1]; zi=pos[3*i+2]; qi=q[i]; }

  {
    int j = jbase + tid;
    float4 t = make_float4(0.f,0.f,0.f,0.f);
    if (j < n) t = make_float4(pos[3*j], pos[3*j+1], pos[3*j+2], q[j]);
    tile[tid] = t;
  }
  __syncthreads();

  const float rc2 = CUTOFF*CUTOFF;
  float acc = 0.f;
  const int cnt = min(256, n - jbase);

  if (diag) {
    const float Lx=cell[0], Ly=cell[4], Lz=cell[8];
    const float iLx=1.f/Lx, iLy=1.f/Ly, iLz=1.f/Lz;
    for (int u = 0; u < cnt; ++u) {
      float4 pj = tile[u];
      int jg = jbase + u;
      float dx = xi - pj.x, dy = yi - pj.y, dz = zi - pj.z;
      dx -= Lx * rintf(dx * iLx);
      dy -= Ly * rintf(dy * iLy);
      dz -= Lz * rintf(dz * iLz);
      float r2 = dx*dx + dy*dy + dz*dz;
      if (i < n && jg != i && r2 < rc2) {
        float r = sqrtf(r2);
        acc += qi * pj.w * erfcf(ALPHA*r) / r;
      }
    }
  } else {
    float inv[3][3]; float det;
    invert3x3(cell, inv, &det);
    float cm[3][3] = {{cell[0],cell[1],cell[2]},
                      {cell[3],cell[4],cell[5]},
                      {cell[6],cell[7],cell[8]}};
    for (int u = 0; u < cnt; ++u) {
      float4 pj = tile[u];
      int jg = jbase + u;
      float dx = xi - pj.x, dy = yi - pj.y, dz = zi - pj.z;
      float fx = dx*inv[0][0] + dy*inv[1][0] + dz*inv[2][0];
      float fy = dx*inv[0][1] + dy*inv[1][1] + dz*inv[2][1];
      float fz = dx*inv[0][2] + dy*inv[1][2] + dz*inv[2][2];
      fx -= rintf(fx); fy -= rintf(fy); fz -= rintf(fz);
      float wx = fx*cm[0][0] + fy*cm[1][0] + fz*cm[2][0];
      float wy = fx*cm[0][1] + fy*cm[1][1] + fz*cm[2][1];
      float wz = fx*cm[0][2] + fy*cm[1][2] + fz*cm[2][2];
      float r2 = wx*wx + wy*wy + wz*wz;
      if (i < n && jg != i && r2 < rc2) {
        float r = sqrtf(r2);
        acc += qi * pj.w * erfcf(ALPHA*r) / r;
      }
    }
  }

  acc *= 0.5f;
  if (i < n && blockIdx.y == 0)
    acc -= (ALPHA * INV_SQRT_PI) * qi * qi;   // self-energy (counted once)

  red[tid] = acc;
  __syncthreads();
  for (int s = 128; s > 0; s >>= 1) {
    if (tid < s) red[tid] += red[tid+s];
    __syncthreads();
  }
  if (tid == 0) atomicAdd(partial, red[0]);
}

// ---------------------------------------------------------------------------
// Reciprocal pass 1: half-space structure factor S(k), k index t in [0,KZERO)
// (Hermitian symmetry: full sum = 2x half sum).  One wave32 handles 16
// k-vectors over one 256-atom chunk staged in LDS.  Phase tiles
// P[16 atoms x 16 kvecs] via V_WMMA_F32_16X16X4_F32 (K padded to 4).
// The B operand carries k/(2*pi), so the WMMA emits phases in *revolutions*
// and feeds v_sin_f32/v_cos_f32 directly (hardware computes sin(2*pi*u)),
// eliminating the per-element 1/(2*pi) pre-scale.
// Inner loop branch-free and software-pipelined:
//   - tpos is float4 {x,y,z,0}: A operand = one aligned ds_load_b64
//   - tq charges: two aligned ds_load_b128 per lane
//   - one zero tile of padding lets us prefetch tile tl+1 unconditionally
// ---------------------------------------------------------------------------
__global__ void ewald_recip_sf(const float* __restrict__ pos,
                               const float* __restrict__ q,
                               const float* __restrict__ cell,
                               float* __restrict__ sre,
                               float* __restrict__ sim, int n) {
  __shared__ float4 tpos[ATOM_CHUNK + 16];   // +1 zero tile for prefetch
  __shared__ float  tq[ATOM_CHUNK + 16];

  const int tid   = threadIdx.x;
  const int lane  = tid & 31;
  const int wave  = tid >> 5;
  const int kbase = (blockIdx.x * 8 + wave) * 16;
  const int abase = blockIdx.y * ATOM_CHUNK;

  // cooperative chunk staging (zero-padded past n and into the pad tile)
  for (int t = tid; t < ATOM_CHUNK + 16; t += 256) {
    int a = abase + t;
    float4 p = make_float4(0.f, 0.f, 0.f, 0.f);
    float  qq = 0.f;
    if (t < ATOM_CHUNK && a < n) {
      p = make_float4(pos[3*a], pos[3*a+1], pos[3*a+2], 0.f);
      qq = q[a];
    }
    tpos[t] = p;
    tq[t]   = qq;
  }
  __syncthreads();

  float inv[3][3]; float det;
  invert3x3(cell, inv, &det);

  // ---- this lane's k-vector column (N = lane & 15), half-space t = j ----
  const int ncol  = lane & 15;
  const int j     = kbase + ncol;
  const bool valid = (j < NKH);
  int tt = valid ? j : 0;
  int nx = tt / (KDIM*KDIM) - KMAX;
  int ny = (tt / KDIM) % KDIM - KMAX;
  int nz = tt % KDIM - KMAX;
  // k/(2*pi): kvec_c/2pi = sum_d n_d * inv_cell[c][d]  (revolution units)
  float kx = (float)nx*inv[0][0] + (float)ny*inv[0][1] + (float)nz*inv[0][2];
  float ky = (float)nx*inv[1][0] + (float)ny*inv[1][1] + (float)nz*inv[1][2];
  float kz = (float)nx*inv[2][0] + (float)ny*inv[2][1] + (float)nz*inv[2][2];
  kx = valid ? kx : 0.f;
  ky = valid ? ky : 0.f;
  kz = valid ? kz : 0.f;

  // ---- B operand (4x16): lanes 0-15 hold K rows {0,1}, lanes 16-31 rows {2,3}
  const bool lo = (lane < 16);
  v2f b;
  b.x = lo ? kx : kz;
  b.y = lo ? ky : 0.f;

  const v8f zeroC = {};
  const int arow = lane & 15;             // A row M = lane % 16
  const int mhi  = (lane >> 4) << 3;      // C/D: lanes 16-31 hold rows M+8
  const int aoff = lo ? 0 : 2;            // float offset into padded float4

  const float* ldsPos = (const float*)tpos;
  const int aFloatBase = (arow << 2) + aoff;          // float index of my A pair
  const int ntiles = (min(ATOM_CHUNK, n - abase) + 15) >> 4;

  float accRe = 0.f, accIm = 0.f;

  // software pipeline: preload tile 0, prefetch tl+1 before the math
  v2f a = *(const v2f*)(ldsPos + aFloatBase);
  const float4* qp = (const float4*)(tq + mhi);
  float4 qa = qp[0], qb = qp[1];

  for (int tl = 0; tl < ntiles; ++tl) {
    // prefetch next tile (pad tile keeps this in-bounds on the last iter)
    v2f an = *(const v2f*)(ldsPos + aFloatBase + (tl + 1) * 64);
    const float4* qpn = (const float4*)(tq + (tl + 1) * 16 + mhi);
    float4 qan = qpn[0], qbn = qpn[1];

    // phase tile (revolutions): 16 atoms x 16 kvecs in one f32 WMMA
    v8f ph = __builtin_amdgcn_wmma_f32_16x16x4_f32(
        /*neg_a=*/false, a, /*neg_b=*/false, b,
        /*c_mod=*/(short)0, zeroC, /*reuse_a=*/false, /*reuse_b=*/false);

    float qm[8] = {qa.x, qa.y, qa.z, qa.w, qb.x, qb.y, qb.z, qb.w};
#pragma unroll
    for (int v = 0; v < 8; ++v) {
      // hardware v_sin/v_cos evaluate sin/cos(2*pi*u): feed revolutions directly
      float s = __builtin_amdgcn_sinf(ph[v]);
      float c = __builtin_amdgcn_cosf(ph[v]);
      accRe += qm[v] * c;
      accIm += qm[v] * s;
    }

    a = an; qa = qan; qb = qbn;
  }

  // fold the two half-columns (lanes L and L^16 hold partial sums of same k)
  accRe += __shfl_xor(accRe, 16, 32);
  accIm += __shfl_xor(accIm, 16, 32);

  if (valid && lane < 16) {
    atomicAdd(&sre[j], accRe);
    atomicAdd(&sim[j], accIm);
  }
}

// ---------------------------------------------------------------------------
// Reciprocal pass 2: E = 2 * (2*pi/V) * sum_{half k} exp(-k^2/4a^2)/k^2 |S|^2
// ---------------------------------------------------------------------------
__global__ void ewald_recip_energy(const float* __restrict__ sre,
                                   const float* __restrict__ sim,
                                   const float* __restrict__ cell,
                                   float* __restrict__ partial) {
  __shared__ float red[256];
  const int tid = threadIdx.x;
  const int j = blockIdx.x * 256 + tid;

  float inv[3][3]; float det;
  invert3x3(cell, inv, &det);
  const float vol = fabsf(det);

  float e = 0.f;
  if (j < NKH) {
    int nx = j / (KDIM*KDIM) - KMAX;
    int ny = (j / KDIM) % KDIM - KMAX;
    int nz = j % KDIM - KMAX;
    float kx = TWO_PI * ((float)nx*inv[0][0] + (float)ny*inv[0][1] + (float)nz*inv[0][2]);
    float ky = TWO_PI * ((float)nx*inv[1][0] + (float)ny*inv[1][1] + (float)nz*inv[1][2]);
    float kz = TWO_PI * ((float)nx*inv[2][0] + (float)ny*inv[2][1] + (float)nz*inv[2][2]);
    float k2 = kx*kx + ky*ky + kz*kz;
    float coeff = expf(-k2 / (4.f*ALPHA*ALPHA)) / k2;
    float re = sre[j], im = sim[j];
    e = 2.f * (TWO_PI / vol) * coeff * (re*re + im*im);
  }

  red[tid] = e;
  __syncthreads();
  for (int s = 128; s > 0; s >>= 1) {
    if (tid < s) red[tid] += red[tid+s];
    __syncthreads();
  }
  if (tid == 0) atomicAdd(partial, red[0]);
}

__global__ void ewald_zero(float* __restrict__ ws, int count) {
  int i = blockIdx.x * 256 + threadIdx.x;
  if (i < count) ws[i] = 0.f;
}

__global__ void ewald_final(const float* __restrict__ ws, float* __restrict__ out) {
  out[0] = ws[0] + ws[1];
}

extern "C" void kernel_launch(void* const* d_in, const int* in_sizes, int n_in,
                              void* d_out, int out_size, void* d_ws, size_t ws_size,
                              hipStream_t stream) {
  const float* pos  = (const float*)d_in[0];
  const float* q    = (const float*)d_in[1];
  const float* cell = (const float*)d_in[2];
  float* out = (float*)d_out;
  float* ws  = (float*)d_ws;
  const int n = in_sizes[1];               // N_ATOMS

  // zero all accumulators (ws is poisoned by the harness)
  ewald_zero<<<(WS_TOTAL + 255)/256, 256, 0, stream>>>(ws, WS_TOTAL);

  // real-space + self
  {
    dim3 grid((n + 255)/256, (n + 255)/256);
    ewald_real<<<grid, 256, 0, stream>>>(pos, q, cell, ws, n);
  }

  // reciprocal: half-space structure factors, then energy (x2 by symmetry)
  {
    int kBlocks = ((NKH + 15)/16 + 7) / 8;         // 8 waves x 16 kvecs per block
    int aChunks = (n + ATOM_CHUNK - 1) / ATOM_CHUNK;
    dim3 grid(kBlocks, aChunks);
    ewald_recip_sf<<<grid, 256, 0, stream>>>(pos, q, cell,
                                             ws + WS_SRE, ws + WS_SIM, n);
    ewald_recip_energy<<<(NKH + 255)/256, 256, 0, stream>>>(ws + WS_SRE, ws + WS_SIM,
                                                            cell, ws + 1);
  }

  ewald_final<<<1, 1, 0, stream>>>(ws, out);
}